// ActVQ_44908178047169
// MI455X (gfx1250) — compile-verified
//
#include <hip/hip_runtime.h>
#include <hip/hip_bf16.h>
#include <math.h>

// ---------------------------------------------------------------------------
// VQ-VAE forward for MI455X (gfx1250, wave32).
// Internal activation layout is channel-last (N, T, 16): one contiguous
// global_load_b64 per WMMA B fragment, b128 stores in the epilogue.
// Each conv maps onto V_WMMA_F32_16X16X4_F32 tiles (fp32 exact):
//   A = 16x4 weight chunk, B = 4x16 activations (16 time steps),
//   D = 16 out-channels x 16 time steps.
// Codebook search: (rows x 16) @ (16 x 2048) WMMA GEMM + argmin, with the
// 128KB codebook staged per-workgroup into LDS via async global->LDS copies
// (codebook exceeds the 64KB WGP$, LDS staging kills 8-way redundant L2 reads).
// ---------------------------------------------------------------------------

typedef __attribute__((ext_vector_type(2))) float v2f;
typedef __attribute__((ext_vector_type(8))) float v8f;

#define CF_PRERELU  1   // relu() applied to loaded activations
#define CF_POSTRELU 2   // relu() applied to stored outputs
#define CF_UPS2     4   // input is nearest-upsampled x2 on the fly
#define CF_IN_NTC4  8   // input tensor is (N,T,4) layout (raw model input)
#define CF_OUT_NTC4 16  // output tensor is (N,T,4) layout (final output)

// Fused conv1d, one wave per (batch, 16-wide time tile), 8 waves per block.
// K = kernel taps (compile time), CHUNKS = Cin/4 (compile time).
template<int K, int CHUNKS>
__global__ __launch_bounds__(256)
void conv1d_wmma(const float* __restrict__ x, const float* __restrict__ w,
                 const float* __restrict__ bias, const float* res, float* out,
                 int Cin, int Cout, int Tin, int Tout,
                 int stride, int pad, int dil, int flags)
{
    const int lane = threadIdx.x & 31;
    const int wave = threadIdx.x >> 5;
    const int half = lane >> 4;          // f32 WMMA K split across wave halves
    const int lm   = lane & 15;
    const int n    = blockIdx.y;
    const int t0   = (blockIdx.x * 8 + wave) << 4;
    if (t0 >= Tout) return;              // wave-uniform: EXEC stays all-ones

    const int tlen  = (flags & CF_UPS2) ? (Tin << 1) : Tin;
    const int kbase = half ? 2 : 0;
    const int ich   = (flags & CF_IN_NTC4) ? 4 : 16;
    const int tt    = t0 + lm;           // this lane's output time (B column)

    v8f acc = {};
#pragma unroll
    for (int kt = 0; kt < K; ++kt) {
        int ti = tt * stride + kt * dil - pad;
        const bool inb = (ti >= 0) && (ti < tlen);
        int tc = inb ? ti : 0;           // clamp: always load in-bounds, mask after
        if (flags & CF_UPS2) tc >>= 1;
        const float* xp = x + ((size_t)n * Tin + tc) * ich;
#pragma unroll
        for (int c = 0; c < CHUNKS; ++c) {
            const int ci = (c << 2) + kbase;
            // A fragment: weights, row m = lm (out channel), K = ci, ci+1
            v2f a;
            a.x = (lm < Cout) ? w[(lm * Cin + ci    ) * K + kt] : 0.f;
            a.y = (lm < Cout) ? w[(lm * Cin + ci + 1) * K + kt] : 0.f;
            // B fragment: contiguous pair of input channels at time tc
            v2f bv = *(const v2f*)(xp + ci);
            if (flags & CF_PRERELU) { bv.x = fmaxf(bv.x, 0.f); bv.y = fmaxf(bv.y, 0.f); }
            if (!inb) { bv.x = 0.f; bv.y = 0.f; }
            acc = __builtin_amdgcn_wmma_f32_16x16x4_f32(
                false, a, false, bv, (short)0, acc, false, false);
        }
    }

    // Epilogue: D lane layout -> column time = tt, rows m = r + 8*half
    float v[8];
    const size_t obase = (size_t)n * Tout + tt;
#pragma unroll
    for (int r = 0; r < 8; ++r) {
        const int m = r + (half ? 8 : 0);
        float t = acc[r] + ((m < Cout) ? bias[m] : 0.f);
        if (m >= Cout) t = 0.f;          // keep pad channels exactly 0
        v[r] = t;
    }
    if (res) {
        const float4* rp = (const float4*)(res + obase * 16 + (half ? 8 : 0));
        float4 r0 = rp[0], r1 = rp[1];
        v[0] += r0.x; v[1] += r0.y; v[2] += r0.z; v[3] += r0.w;
        v[4] += r1.x; v[5] += r1.y; v[6] += r1.z; v[7] += r1.w;
    }
    if (flags & CF_POSTRELU) {
#pragma unroll
        for (int r = 0; r < 8; ++r) v[r] = fmaxf(v[r], 0.f);
    }
    if (flags & CF_OUT_NTC4) {
        if (half == 0)                   // rows 0..3 are the 4 real channels
            *(float4*)(out + obase * 4) = make_float4(v[0], v[1], v[2], v[3]);
    } else {
        float4* op = (float4*)(out + obase * 16 + (half ? 8 : 0));
        op[0] = make_float4(v[0], v[1], v[2], v[3]);
        op[1] = make_float4(v[4], v[5], v[6], v[7]);
    }
}

// ---------------- vector quantization -------------------------------------

__global__ void vq_init(int* counts, float* commit, int nb)
{
    int i = blockIdx.x * blockDim.x + threadIdx.x;
    if (i < nb) counts[i] = 0;
    if (i == 0) commit[0] = 0.f;
}

__global__ void vq_cnorm(const float* __restrict__ cb, float* cnorm, int nb)
{
    int i = blockIdx.x * blockDim.x + threadIdx.x;
    if (i < nb) {
        float s = 0.f;
#pragma unroll
        for (int c4 = 0; c4 < 4; ++c4) {
            float4 q = ((const float4*)(cb + (size_t)i * 16))[c4];
            s += q.x * q.x + q.y * q.y + q.z * q.z + q.w * q.w;
        }
        cnorm[i] = s;
    }
}

// One wave handles 16 rows; loops over all codes in 16-wide tiles fed from LDS.
// score = |code|^2 - 2 * <x, code>  (row norm constant under argmin)
extern __shared__ float vq_smem[];   // [nb*16] codebook + [nb] code norms

__global__ __launch_bounds__(256)
void vq_argmin(const float* __restrict__ enc, const float* __restrict__ cb,
               const float* __restrict__ cnorm, int* idx, int nb)
{
    // ---- stage codebook + norms into LDS with async global->LDS copies ----
    {
        const int tid = threadIdx.x;
        for (int i = tid; i < nb * 4; i += 256) {          // nb*4 float4 tiles
            unsigned long long g = (unsigned long long)(size_t)(cb + (size_t)i * 4);
            unsigned l = (unsigned)(size_t)(vq_smem + (size_t)i * 4);
            asm volatile("global_load_async_to_lds_b128 %0, %1, off"
                         :: "v"(l), "v"(g) : "memory");
        }
        for (int i = tid; i < nb / 4; i += 256) {          // cnorm float4 tiles
            unsigned long long g = (unsigned long long)(size_t)(cnorm + (size_t)i * 4);
            unsigned l = (unsigned)(size_t)(vq_smem + (size_t)nb * 16 + (size_t)i * 4);
            asm volatile("global_load_async_to_lds_b128 %0, %1, off"
                         :: "v"(l), "v"(g) : "memory");
        }
        asm volatile("s_wait_asynccnt 0" ::: "memory");
    }
    __syncthreads();

    const int lane = threadIdx.x & 31;
    const int wave = threadIdx.x >> 5;
    const int half = lane >> 4;
    const int lm   = lane & 15;
    const int kbase = half ? 2 : 0;
    const int rowBase = (blockIdx.x * 8 + wave) << 4;

    // A fragments: 16 rows x 16 dims (4 chunks of K=4), contiguous b64 loads
    const int row = rowBase + lm;
    v2f a[4];
#pragma unroll
    for (int c = 0; c < 4; ++c)
        a[c] = *(const v2f*)(enc + (size_t)row * 16 + (c << 2) + kbase);

    float best[8]; int bidx[8];
#pragma unroll
    for (int r = 0; r < 8; ++r) { best[r] = 3.4e38f; bidx[r] = 0; }

    for (int cb0 = 0; cb0 < nb; cb0 += 16) {
        const int code = cb0 + lm;                  // this lane's D column
        v8f acc = {};
#pragma unroll
        for (int c = 0; c < 4; ++c) {
            v2f b = *(const v2f*)(vq_smem + (size_t)code * 16 + (c << 2) + kbase);
            acc = __builtin_amdgcn_wmma_f32_16x16x4_f32(
                false, a[c], false, b, (short)0, acc, false, false);
        }
        const float cn = vq_smem[(size_t)nb * 16 + code];
#pragma unroll
        for (int r = 0; r < 8; ++r) {
            float s = cn - 2.f * acc[r];
            if (s < best[r]) { best[r] = s; bidx[r] = code; }
        }
    }

    // Reduce (min, lowest-index) across the 16 lanes of each half-wave.
#pragma unroll
    for (int r = 0; r < 8; ++r) {
        float b0 = best[r]; int i0 = bidx[r];
#pragma unroll
        for (int off = 1; off < 16; off <<= 1) {
            float ob = __shfl_xor(b0, off, 32);
            int   oi = __shfl_xor(i0, off, 32);
            if (ob < b0 || (ob == b0 && oi < i0)) { b0 = ob; i0 = oi; }
        }
        if (lm == 0) idx[rowBase + r + (half ? 8 : 0)] = i0;
    }
}

__global__ void vq_gather(const float* __restrict__ enc, const float* __restrict__ cb,
                          const int* __restrict__ idx, float* xq,
                          float* commit, int* counts, int rows)
{
    const int row = blockIdx.x * blockDim.x + threadIdx.x;
    float s = 0.f;
    if (row < rows) {
        const int code = idx[row];
        atomicAdd(&counts[code], 1);
#pragma unroll
        for (int c4 = 0; c4 < 4; ++c4) {
            float4 f = ((const float4*)(enc + (size_t)row * 16))[c4];
            float4 q = ((const float4*)(cb + (size_t)code * 16))[c4];
            ((float4*)(xq + (size_t)row * 16))[c4] = q;
            float dx = f.x - q.x, dy = f.y - q.y, dz = f.z - q.z, dw = f.w - q.w;
            s += dx * dx + dy * dy + dz * dz + dw * dw;
        }
    }
    __shared__ float sh[256];
    sh[threadIdx.x] = s; __syncthreads();
    for (int o = 128; o; o >>= 1) {
        if (threadIdx.x < o) sh[threadIdx.x] += sh[threadIdx.x + o];
        __syncthreads();
    }
    if (threadIdx.x == 0) atomicAdd(commit, sh[0]);
}

__global__ void vq_finalize(const float* commit, const int* counts,
                            float* outTail, int rows, int nb)
{
    __shared__ float sh[256];
    float s = 0.f;
    for (int i = threadIdx.x; i < nb; i += 256) {
        float p = (float)counts[i] / (float)rows;
        s += p * logf(p + 1e-7f);
    }
    sh[threadIdx.x] = s; __syncthreads();
    for (int o = 128; o; o >>= 1) {
        if (threadIdx.x < o) sh[threadIdx.x] += sh[threadIdx.x + o];
        __syncthreads();
    }
    if (threadIdx.x == 0) {
        outTail[0] = commit[0] / ((float)rows * 16.f);   // commit loss (mean)
        outTail[1] = expf(-sh[0]);                       // perplexity
    }
}

// ---------------------------------------------------------------------------

extern "C" void kernel_launch(void* const* d_in, const int* in_sizes, int n_in,
                              void* d_out, int out_size, void* d_ws, size_t ws_size,
                              hipStream_t stream)
{
    (void)in_sizes; (void)n_in; (void)out_size; (void)ws_size;

    const int N = 64, Tfull = 4096, Tq = 512, NB = 2048;
    const size_t BIG = (size_t)N * Tfull * 16;   // 4,194,304 floats
    const size_t ENC = (size_t)N * Tq * 16;      //   524,288 floats

    float* A  = (float*)d_ws;
    float* Bb = A + BIG;
    float* E  = Bb + BIG;                        // encoder output (N,512,16)
    float* Q  = E + ENC;                         // quantized (N,512,16)
    float* cn = Q + ENC;                         // code norms (2048)
    float* cm = cn + NB;                         // commit accumulator
    int*   ix = (int*)(cm + 16);                 // argmin indices (32768)
    int*   ct = ix + N * Tq;                     // code counts (2048)
    // total ws use: ~38 MB

    const float* X  = (const float*)d_in[0];
    const float* CB = (const float*)d_in[95];
    auto F = [&](int i) { return (const float*)d_in[i]; };

    auto conv = [&](const float* x, int wi, int bi, const float* res, float* out,
                    int Cin, int Cout, int Tin, int Tout,
                    int K, int stride, int pad, int dil, int flags) {
        dim3 grid((Tout / 16 + 7) / 8, N);
        if (K == 3 && Cin == 4)
            conv1d_wmma<3, 1><<<grid, 256, 0, stream>>>(x, F(wi), F(bi), res, out,
                Cin, Cout, Tin, Tout, stride, pad, dil, flags);
        else if (K == 3)
            conv1d_wmma<3, 4><<<grid, 256, 0, stream>>>(x, F(wi), F(bi), res, out,
                Cin, Cout, Tin, Tout, stride, pad, dil, flags);
        else if (K == 4)
            conv1d_wmma<4, 4><<<grid, 256, 0, stream>>>(x, F(wi), F(bi), res, out,
                Cin, Cout, Tin, Tout, stride, pad, dil, flags);
        else
            conv1d_wmma<1, 4><<<grid, 256, 0, stream>>>(x, F(wi), F(bi), res, out,
                Cin, Cout, Tin, Tout, stride, pad, dil, flags);
    };

    // ---------------- encoder ----------------
    // h = relu(conv(x, k3 p1)), input is (N,T,4)
    conv(X, 1, 2, nullptr, A, 4, 16, Tfull, Tfull, 3, 1, 1, 1, CF_IN_NTC4 | CF_POSTRELU);
    float* cur = A; float* tmp = Bb;
    int T = Tfull;
    int p = 3;
    for (int blk = 0; blk < 3; ++blk) {
        // down conv: k4 s2 p1
        conv(cur, p, p + 1, nullptr, tmp, 16, 16, T, T / 2, 4, 2, 1, 1, 0);
        p += 2; T /= 2;
        { float* t2 = cur; cur = tmp; tmp = t2; }
        for (int j = 0; j < 3; ++j) {
            const int dil = 1 << j;   // dilation_growth_rate=2, forward order
            conv(cur, p, p + 1, nullptr, tmp, 16, 16, T, T, 3, 1, dil, dil, CF_PRERELU);
            conv(tmp, p + 2, p + 3, cur, cur, 16, 16, T, T, 1, 1, 0, 1, CF_PRERELU);
            p += 4;
        }
    }
    // encoder out conv: k3 p1 (p == 45)
    conv(cur, 45, 46, nullptr, E, 16, 16, Tq, Tq, 3, 1, 1, 1, 0);

    // ---------------- quantize ----------------
    vq_init  <<<(NB + 255) / 256, 256, 0, stream>>>(ct, cm, NB);
    vq_cnorm <<<(NB + 255) / 256, 256, 0, stream>>>(CB, cn, NB);
    {
        const size_t smemBytes = (size_t)(NB * 16 + NB) * sizeof(float); // 136 KB
        vq_argmin<<<(N * Tq) / 128, 256, smemBytes, stream>>>(E, CB, cn, ix, NB);
    }
    vq_gather<<<(N * Tq) / 256, 256, 0, stream>>>(E, CB, ix, Q, cm, ct, N * Tq);

    // ---------------- decoder ----------------
    conv(Q, 47, 48, nullptr, A, 16, 16, Tq, Tq, 3, 1, 1, 1, CF_POSTRELU);
    cur = A; tmp = Bb; T = Tq; p = 49;
    for (int blk = 0; blk < 3; ++blk) {
        for (int j = 0; j < 3; ++j) {
            const int dil = 1 << (2 - j);   // reverse_dilation=True
            conv(cur, p, p + 1, nullptr, tmp, 16, 16, T, T, 3, 1, dil, dil, CF_PRERELU);
            conv(tmp, p + 2, p + 3, cur, cur, 16, 16, T, T, 1, 1, 0, 1, CF_PRERELU);
            p += 4;
        }
        // nearest upsample x2 fused into conv k3 p1
        conv(cur, p, p + 1, nullptr, tmp, 16, 16, T, 2 * T, 3, 1, 1, 1, CF_UPS2);
        p += 2; T *= 2;
        { float* t2 = cur; cur = tmp; tmp = t2; }
    }
    // mid conv + relu (p == 91)
    conv(cur, 91, 92, nullptr, tmp, 16, 16, Tfull, Tfull, 3, 1, 1, 1, CF_POSTRELU);
    { float* t2 = cur; cur = tmp; tmp = t2; }
    // out conv -> (N,T,4) directly into d_out
    conv(cur, 93, 94, nullptr, (float*)d_out, 16, 4, Tfull, Tfull, 3, 1, 1, 1, CF_OUT_NTC4);

    // scalar tail: d_out[1048576] = commit loss, d_out[1048577] = perplexity
    vq_finalize<<<1, 256, 0, stream>>>(cm, ct, (float*)d_out + (size_t)N * Tfull * 4,
                                       N * Tq, NB);
}